// BA_Shapes_GCN_edge_26371099198063
// MI455X (gfx1250) — compile-verified
//
#include <hip/hip_runtime.h>

// ---------------------------------------------------------------------------
// GCN link-prediction pipeline for MI455X (gfx1250, wave32, WMMA).
//
// Sizes (from reference):
//   N_NODES=50000, N_EDGES=800000, F=128, edge rows = 2*100000 + 2*20000 = 240000
//
// All GEMMs use V_WMMA_F32_16X16X4_F32 (fp32 in/out -> bit-compatible precision
// with the fp32 reference up to reduction order). One wave owns a 16x128
// output strip: 8x v8f accumulators, K stepped by 4.
// ---------------------------------------------------------------------------

typedef float v2f __attribute__((ext_vector_type(2)));
typedef float v8f __attribute__((ext_vector_type(8)));

#define N_NODES 50000
#define N_EDGES 800000
#define FDIM    128
#define E_TR    100000
#define E_TE    20000
#define N_ROWS  240000   // 2*E_TR + 2*E_TE

// ---------------------------------------------------------------------------
// Degree / normalization
// ---------------------------------------------------------------------------
__global__ __launch_bounds__(256) void deg_count(const int* __restrict__ dst,
                                                 float* __restrict__ deg, int nE) {
    int t = blockIdx.x * blockDim.x + threadIdx.x;
    if (t < nE) atomicAdd(&deg[dst[t]], 1.0f);
}

__global__ __launch_bounds__(256) void make_dis(float* __restrict__ deg, int n) {
    int t = blockIdx.x * blockDim.x + threadIdx.x;
    if (t < n) deg[t] = rsqrtf(deg[t] + 1.0f);   // dis = (deg+1)^-0.5, in place
}

// ---------------------------------------------------------------------------
// Node GEMM: C[M x 128] = A[M x 128] @ B[128 x 128]  (row-major, no bias)
// One wave -> 16 rows x 128 cols. 8 waves per block.
// WMMA f32 16x16x4 layout:
//   A frag (16x4): lane L holds row (L%16), K = k0 + 2*(L/16) + {0,1}
//   B frag (4x16): lane L holds col (L%16), K = k0 + 2*(L/16) + {0,1}
//   C tile (16x16): VGPR g, lanes 0-15 -> row g, lanes 16-31 -> row g+8
// ---------------------------------------------------------------------------
__global__ __launch_bounds__(256) void gemm_node(const float* __restrict__ A,
                                                 const float* __restrict__ B,
                                                 float* __restrict__ C, int M) {
    int wave = (blockIdx.x * blockDim.x + threadIdx.x) >> 5;
    int lane = threadIdx.x & 31;
    int m0 = wave * 16;
    if (m0 >= M) return;                  // wave-uniform: EXEC stays all-1s
    int half = lane >> 4;                 // 0 or 1
    int l16  = lane & 15;

    const float* Arow = A + (size_t)(m0 + l16) * FDIM;

    v8f acc[8] = {};                      // 16x128 strip of C

    for (int k0 = 0; k0 < FDIM; k0 += 4) {
        int k = k0 + half * 2;
        v2f afrag = *(const v2f*)(Arow + k);   // K=k, K=k+1 (8B aligned)
        const float* b0 = B + (size_t)k * FDIM + l16;
        #pragma unroll
        for (int n = 0; n < 8; ++n) {
            v2f bfrag;
            bfrag.x = b0[n * 16];
            bfrag.y = b0[FDIM + n * 16];
            acc[n] = __builtin_amdgcn_wmma_f32_16x16x4_f32(
                false, afrag, false, bfrag, (short)0, acc[n], false, false);
        }
    }

    #pragma unroll
    for (int n = 0; n < 8; ++n) {
        #pragma unroll
        for (int g = 0; g < 8; ++g) {
            int m = m0 + g + half * 8;
            C[(size_t)m * FDIM + n * 16 + l16] = acc[n][g];
        }
    }
}

// ---------------------------------------------------------------------------
// Edge scatter:  agg[dst] += hw[src] * dis[src]   (one wave per edge)
// src/dst uniform per wave -> fully coalesced 512B gather, L2 f32 atomics.
// ---------------------------------------------------------------------------
__global__ __launch_bounds__(256) void scatter_add(const float* __restrict__ hw,
                                                   const float* __restrict__ dis,
                                                   const int* __restrict__ src,
                                                   const int* __restrict__ dst,
                                                   float* __restrict__ agg, int nE) {
    int t = blockIdx.x * blockDim.x + threadIdx.x;
    int e = t >> 5;
    if (e >= nE) return;
    int f = (t & 31) * 4;
    int s = src[e], d = dst[e];
    float sc = dis[s];
    const float4 v = *(const float4*)(hw + (size_t)s * FDIM + f);
    float* dp = agg + (size_t)d * FDIM + f;
    atomicAdd(dp + 0, v.x * sc);
    atomicAdd(dp + 1, v.y * sc);
    atomicAdd(dp + 2, v.z * sc);
    atomicAdd(dp + 3, v.w * sc);
}

// ---------------------------------------------------------------------------
// Combine:  h_out = [relu]( agg*dis + hw*dis^2 + b )
// ---------------------------------------------------------------------------
__global__ __launch_bounds__(256) void combine(const float* __restrict__ agg,
                                               const float* __restrict__ hw,
                                               const float* __restrict__ dis,
                                               const float* __restrict__ b,
                                               float* __restrict__ hout,
                                               int nNodes, int doRelu) {
    int t = blockIdx.x * blockDim.x + threadIdx.x;
    int node = t >> 5;
    if (node >= nNodes) return;
    int f = (t & 31) * 4;
    float di  = dis[node];
    float di2 = di * di;
    float4 a = *(const float4*)(agg + (size_t)node * FDIM + f);
    float4 h = *(const float4*)(hw  + (size_t)node * FDIM + f);
    float4 r;
    r.x = a.x * di + h.x * di2 + b[f + 0];
    r.y = a.y * di + h.y * di2 + b[f + 1];
    r.z = a.z * di + h.z * di2 + b[f + 2];
    r.w = a.w * di + h.w * di2 + b[f + 3];
    if (doRelu) {
        r.x = fmaxf(r.x, 0.0f); r.y = fmaxf(r.y, 0.0f);
        r.z = fmaxf(r.z, 0.0f); r.w = fmaxf(r.w, 0.0f);
    }
    *(float4*)(hout + (size_t)node * FDIM + f) = r;
}

// ---------------------------------------------------------------------------
// Fused edge MLP:
//   row r -> (u,v) edge, A-row = concat(h[u], h[v])  (K=256)
//   z = relu(A @ Wl1 + bl1)  via WMMA f32 16x16x4 (one wave = 16 rows)
//   out[r] = dot(z, Wl2) + bl2  via in-register accumulate + 16-lane reduction
// ---------------------------------------------------------------------------
__global__ __launch_bounds__(256) void gemm_edge(const float* __restrict__ H,
                                                 const int* __restrict__ pos_tr,
                                                 const int* __restrict__ neg_tr,
                                                 const int* __restrict__ pos_te,
                                                 const int* __restrict__ neg_te,
                                                 const float* __restrict__ Wl1,
                                                 const float* __restrict__ bl1,
                                                 const float* __restrict__ Wl2,
                                                 const float* __restrict__ bl2,
                                                 float* __restrict__ out) {
    int wave = (blockIdx.x * blockDim.x + threadIdx.x) >> 5;
    int lane = threadIdx.x & 31;
    int r0 = wave * 16;
    if (r0 >= N_ROWS) return;             // wave-uniform
    int half = lane >> 4;
    int l16  = lane & 15;
    int r = r0 + l16;

    // resolve (u, v) for this lane's A-row (train pos, train neg, test pos, test neg)
    int u, v;
    if (r < E_TR)            { u = pos_tr[r];              v = pos_tr[E_TR + r]; }
    else if (r < 2 * E_TR)   { int i = r - E_TR;           u = neg_tr[i]; v = neg_tr[E_TR + i]; }
    else if (r < 2 * E_TR + E_TE) { int i = r - 2 * E_TR;  u = pos_te[i]; v = pos_te[E_TE + i]; }
    else                     { int i = r - 2 * E_TR - E_TE; u = neg_te[i]; v = neg_te[E_TE + i]; }

    const float* Au = H + (size_t)u * FDIM;
    const float* Av = H + (size_t)v * FDIM;

    v8f acc[8] = {};

    for (int k0 = 0; k0 < 2 * FDIM; k0 += 4) {
        int k = k0 + half * 2;            // k, k+1 never straddle the 128 boundary
        const float* ap = (k < FDIM) ? (Au + k) : (Av + (k - FDIM));
        v2f afrag = *(const v2f*)ap;
        const float* b0 = Wl1 + (size_t)k * FDIM + l16;
        #pragma unroll
        for (int n = 0; n < 8; ++n) {
            v2f bfrag;
            bfrag.x = b0[n * 16];
            bfrag.y = b0[FDIM + n * 16];
            acc[n] = __builtin_amdgcn_wmma_f32_16x16x4_f32(
                false, afrag, false, bfrag, (short)0, acc[n], false, false);
        }
    }

    // epilogue: +bias, relu, multiply by Wl2 column, accumulate per output row
    float partial[8];
    #pragma unroll
    for (int g = 0; g < 8; ++g) partial[g] = 0.0f;

    #pragma unroll
    for (int n = 0; n < 8; ++n) {
        float bb = bl1[n * 16 + l16];
        float ww = Wl2[n * 16 + l16];
        #pragma unroll
        for (int g = 0; g < 8; ++g) {
            float z = fmaxf(acc[n][g] + bb, 0.0f);
            partial[g] += z * ww;
        }
    }

    float bias2 = bl2[0];
    #pragma unroll
    for (int g = 0; g < 8; ++g) {
        float s = partial[g];
        // reduce across the 16 lanes of this half-wave (row = g + half*8)
        s += __shfl_xor(s, 8, 16);
        s += __shfl_xor(s, 4, 16);
        s += __shfl_xor(s, 2, 16);
        s += __shfl_xor(s, 1, 16);
        if (l16 == 0) out[r0 + g + half * 8] = s + bias2;
    }
}

// ---------------------------------------------------------------------------
// Host orchestration (graph-capture safe: only async ops on `stream`)
// ---------------------------------------------------------------------------
extern "C" void kernel_launch(void* const* d_in, const int* in_sizes, int n_in,
                              void* d_out, int out_size, void* d_ws, size_t ws_size,
                              hipStream_t stream) {
    const float* x          = (const float*)d_in[0];
    const int*   edge_index = (const int*)d_in[1];
    const int*   pos_tr     = (const int*)d_in[2];
    const int*   neg_tr     = (const int*)d_in[3];
    const int*   pos_te     = (const int*)d_in[4];
    const int*   neg_te     = (const int*)d_in[5];
    const float* W[4] = {(const float*)d_in[6],  (const float*)d_in[8],
                         (const float*)d_in[10], (const float*)d_in[12]};
    const float* b[4] = {(const float*)d_in[7],  (const float*)d_in[9],
                         (const float*)d_in[11], (const float*)d_in[13]};
    const float* Wl1 = (const float*)d_in[14];
    const float* bl1 = (const float*)d_in[15];
    const float* Wl2 = (const float*)d_in[16];
    const float* bl2 = (const float*)d_in[17];
    float* out = (float*)d_out;

    const size_t NF = (size_t)N_NODES * FDIM;   // 6,400,000 floats
    float* ws  = (float*)d_ws;
    float* dis = ws;                 // 50000 (padded to 50048 for alignment)
    float* hw  = ws + 50048;         // N_NODES x 128
    float* agg = hw  + NF;           // N_NODES x 128
    float* h0  = agg + NF;           // ping
    float* h1  = h0  + NF;           // pong

    const int* src = edge_index;
    const int* dst = edge_index + N_EDGES;

    // --- degree / dis ---
    hipMemsetAsync(dis, 0, N_NODES * sizeof(float), stream);
    deg_count<<<(N_EDGES + 255) / 256, 256, 0, stream>>>(dst, dis, N_EDGES);
    make_dis<<<(N_NODES + 255) / 256, 256, 0, stream>>>(dis, N_NODES);

    // --- 4 GCN layers (ReLU on 0..2) ---
    const int tilesM     = (N_NODES + 15) / 16;               // 3125
    const int gemmBlocks = (tilesM + 7) / 8;                  // 8 waves / block
    const float* hin = x;
    float* houts[4] = {h0, h1, h0, h1};
    for (int l = 0; l < 4; ++l) {
        gemm_node<<<gemmBlocks, 256, 0, stream>>>(hin, W[l], hw, N_NODES);
        hipMemsetAsync(agg, 0, NF * sizeof(float), stream);
        scatter_add<<<(N_EDGES * 32) / 256, 256, 0, stream>>>(hw, dis, src, dst, agg, N_EDGES);
        combine<<<(N_NODES * 32 + 255) / 256, 256, 0, stream>>>(
            agg, hw, dis, b[l], houts[l], N_NODES, (l < 3) ? 1 : 0);
        hin = houts[l];
    }

    // --- fused edge MLP -> 240000 logits (train rows first, then test) ---
    const int edgeTiles  = N_ROWS / 16;                       // 15000
    const int edgeBlocks = (edgeTiles + 7) / 8;               // 1875
    gemm_edge<<<edgeBlocks, 256, 0, stream>>>(h1, pos_tr, neg_tr, pos_te, neg_te,
                                              Wl1, bl1, Wl2, bl2, out);
}